// PositionalEncodingLayer_44633300140770
// MI455X (gfx1250) — compile-verified
//
#include <hip/hip_runtime.h>
#include <cstdint>
#include <cstddef>

// Positional encoding add: out[b,s,d] = X[b,s,d] + PE(s,d)
//   PE(s,2i)   = sin(s / 10000^(2i/D))
//   PE(s,2i+1) = cos(s / 10000^(2i/D))
// X: [4, 4096, 2048] fp32. Pure streaming op -> async global->LDS DMA path,
// double buffered per wave, NT 128-bit stores.

#define EMBED_D   2048
#define SEQ_S     4096
#define COLS      (EMBED_D / 4)     // 512 float4 per row
#define RPB       8                 // rows per block
#define BLOCK     COLS              // 512 threads, 16 waves

typedef float v4f __attribute__((ext_vector_type(4)));

__device__ __forceinline__ void async_load_b128(const v4f* gptr, uint32_t lds_byte_addr) {
    // GLOBAL_LOAD_ASYNC_TO_LDS_B128: vdst = LDS byte address (VGPR),
    // vaddr = 64-bit global address. Tracked by ASYNCcnt.
    asm volatile("global_load_async_to_lds_b128 %0, %1, off"
                 :
                 : "v"(lds_byte_addr), "v"(gptr)
                 : "memory");
}

__device__ __forceinline__ void wait_async_le1() {
    asm volatile("s_wait_asynccnt 0x1" ::: "memory");
}
__device__ __forceinline__ void wait_async_le0() {
    asm volatile("s_wait_asynccnt 0x0" ::: "memory");
}
__device__ __forceinline__ void wait_ds0() {
    asm volatile("s_wait_dscnt 0x0" ::: "memory");
}

__global__ __launch_bounds__(BLOCK) void
PositionalEncodingLayer_44633300140770_kernel(const float* __restrict__ X,
                                              float* __restrict__ O) {
    __shared__ v4f lbuf[2 * COLS];  // 16 KiB double buffer

    const int col  = threadIdx.x;             // 0..511, fixed per thread
    const int row0 = blockIdx.x * RPB;        // first global row of this block

    // Per-thread inverse frequencies (computed once).
    // float4 covers dims 4c..4c+3 -> sin/cos pairs i0 = 2c, i1 = 2c+1.
    // inv_freq(i) = 10000^(-2i/D) = exp2(-i * (2/D) * log2(10000))
    const float kexp = -0.0129762816206538f;  // -(2/2048)*log2(10000)
    const float inv0 = exp2f(kexp * (float)(2 * col));
    const float inv1 = exp2f(kexp * (float)(2 * col + 1));

    const v4f* Xg = (const v4f*)X + (size_t)row0 * COLS + col;
    v4f*       Og = (v4f*)O       + (size_t)row0 * COLS + col;

    // Hardware LDS byte addresses for this lane's two buffer slots.
    // Generic shared-pointer low 32 bits == LDS byte offset (ISA §10.2).
    const uint32_t lds0 = (uint32_t)(uintptr_t)&lbuf[col];
    const uint32_t lds1 = (uint32_t)(uintptr_t)&lbuf[COLS + col];

    // Prologue: prime both buffers (ASYNCcnt = 2).
    async_load_b128(Xg,        lds0);
    async_load_b128(Xg + COLS, lds1);

#pragma unroll
    for (int r = 0; r < RPB; ++r) {
        // Wait for the oldest async load (row r) to land in LDS.
        if (r + 1 < RPB) wait_async_le1();
        else             wait_async_le0();

        // Consume own slot: ds_load_b128, no cross-lane sharing -> no barrier.
        const v4f x = lbuf[(r & 1) * COLS + col];

        const float pos = (float)((row0 + r) & (SEQ_S - 1));
        const float a0  = pos * inv0;
        const float a1  = pos * inv1;

        v4f o;
        o.x = x.x + __sinf(a0);
        o.y = x.y + __cosf(a0);
        o.z = x.z + __sinf(a1);
        o.w = x.w + __cosf(a1);

        // Streamed output, no reuse -> non-temporal 128-bit store.
        __builtin_nontemporal_store(o, Og + (size_t)r * COLS);

        // Refill this buffer with row r+2. Close the LDS read-vs-async-write
        // race: ensure the ds_load above fully completed first.
        if (r + 2 < RPB) {
            wait_ds0();
            async_load_b128(Xg + (size_t)(r + 2) * COLS,
                            (r & 1) ? lds1 : lds0);
        }
    }
}

extern "C" void kernel_launch(void* const* d_in, const int* in_sizes, int n_in,
                              void* d_out, int out_size, void* d_ws, size_t ws_size,
                              hipStream_t stream) {
    (void)n_in; (void)out_size; (void)d_ws; (void)ws_size;
    const float* X = (const float*)d_in[0];
    float*       O = (float*)d_out;

    const int n_elems = in_sizes[0];          // 4*4096*2048 = 33,554,432
    const int rows    = n_elems / EMBED_D;    // 16384
    const int grid    = rows / RPB;           // 2048 blocks x 512 threads

    PositionalEncodingLayer_44633300140770_kernel<<<dim3(grid), dim3(BLOCK), 0, stream>>>(X, O);
}